// BochnerKernel_46411416601270
// MI455X (gfx1250) — compile-verified
//
#include <hip/hip_runtime.h>

typedef float v2f __attribute__((ext_vector_type(2)));
typedef float v8f __attribute__((ext_vector_type(8)));

#define NROWS 8192
#define MROWS 8192
#define DDIM  256
#define KC    32
#define LDK   (KC + 4)   // 36: rows 16B-aligned AND 36*r mod 64 distinct for r=0..15

// ---------------------------------------------------------------------------
// Stage 1: per-row precompute.  For each row r of Xin (length 256):
//   proj[r][k]   = sum_d Xin[r][d] * W[d][k]        (row @ W)
//   scaled[r][k] = Xin[r][k] / ls[k]
//   nProj[r]     = ||proj[r]||^2
//   nScaled[r]   = ||scaled[r]||^2
//   rb[r]        = row . b
// ---------------------------------------------------------------------------
__global__ __launch_bounds__(256) void precompute_side(
    const float* __restrict__ Xin, const float* __restrict__ W,
    const float* __restrict__ b,   const float* __restrict__ ls,
    float* __restrict__ proj, float* __restrict__ scaled,
    float* __restrict__ nProj, float* __restrict__ nScaled,
    float* __restrict__ rb)
{
    __shared__ float row[DDIM];
    __shared__ float red[256];
    const int r = blockIdx.x;
    const int t = threadIdx.x;

    const float xv = Xin[(size_t)r * DDIM + t];
    row[t] = xv;
    __syncthreads();

    float acc = 0.f;
#pragma unroll 8
    for (int d = 0; d < DDIM; ++d)
        acc = fmaf(row[d], W[d * DDIM + t], acc);
    proj[(size_t)r * DDIM + t] = acc;

    const float sv = xv / ls[t];
    scaled[(size_t)r * DDIM + t] = sv;

    // reduction 1: row . b
    red[t] = xv * b[t];
    __syncthreads();
    for (int s = 128; s > 0; s >>= 1) {
        if (t < s) red[t] += red[t + s];
        __syncthreads();
    }
    if (t == 0) rb[r] = red[0];
    __syncthreads();

    // reduction 2: ||proj||^2
    red[t] = acc * acc;
    __syncthreads();
    for (int s = 128; s > 0; s >>= 1) {
        if (t < s) red[t] += red[t + s];
        __syncthreads();
    }
    if (t == 0) nProj[r] = red[0];
    __syncthreads();

    // reduction 3: ||scaled||^2
    red[t] = sv * sv;
    __syncthreads();
    for (int s = 128; s > 0; s >>= 1) {
        if (t < s) red[t] += red[t + s];
        __syncthreads();
    }
    if (t == 0) nScaled[r] = red[0];
}

// ---------------------------------------------------------------------------
// Stage 2: fused pairwise kernel.
// Workgroup = 256 threads = 8 waves, output tile 128 (i) x 128 (j).
// Wave grid 4 (i) x 2 (j); each wave owns a 32x64 subtile:
//   2x4 fp32 WMMA C tiles for the adaptive GEMM (A@B^T) and 2x4 for the
//   gaussian GEMM (Sx@Sy^T)  -> 16 v8f accumulators, 12 b64 frag loads
//   per k-step feeding 16 v_wmma_f32_16x16x4_f32.
// LDS rows padded to 36 dwords: 16B-aligned b128 fills + conflict-free
// 8B-aligned ds_load_b64 fragment reads.
// ---------------------------------------------------------------------------
__global__ __launch_bounds__(256) void bochner_main(
    const float* __restrict__ A,   // X@W   [N,D]
    const float* __restrict__ B,   // Y@W   [M,D]
    const float* __restrict__ Sx,  // X/ls  [N,D]
    const float* __restrict__ Sy,  // Y/ls  [M,D]
    const float* __restrict__ nA,  const float* __restrict__ nB,
    const float* __restrict__ nSx, const float* __restrict__ nSy,
    const float* __restrict__ xb,  const float* __restrict__ yb,
    float* __restrict__ out)
{
    __shared__ __align__(16) float sAa[128 * LDK];   // adaptive, i-side
    __shared__ __align__(16) float sAg[128 * LDK];   // gaussian, i-side
    __shared__ __align__(16) float sBa[128 * LDK];   // adaptive, j-side
    __shared__ __align__(16) float sBg[128 * LDK];   // gaussian, j-side
    __shared__ float s_nA[128], s_nSx[128], s_xb[128];
    __shared__ float s_nB[128], s_nSy[128], s_yb[128];

    const int tid  = threadIdx.x;
    const int wave = tid >> 5;
    const int lane = tid & 31;
    const int half = lane >> 4;     // lane half selects K pair / M+8
    const int lj   = lane & 15;
    const int wi   = wave >> 1;     // 0..3 : i sub-block of 32
    const int wj   = wave & 1;      // 0..1 : j sub-block of 64
    const int i0   = blockIdx.y * 128;
    const int j0   = blockIdx.x * 128;

    if (tid < 128) {
        s_nA[tid] = nA[i0 + tid];  s_nSx[tid] = nSx[i0 + tid]; s_xb[tid] = xb[i0 + tid];
        s_nB[tid] = nB[j0 + tid];  s_nSy[tid] = nSy[j0 + tid]; s_yb[tid] = yb[j0 + tid];
    }

    v8f cA[2][4] = {};
    v8f cG[2][4] = {};

    for (int k0 = 0; k0 < DDIM; k0 += KC) {
        __syncthreads();
        // fill all four 128 x KC tiles with b128 loads/stores
        for (int idx = tid; idx < 128 * (KC / 4); idx += 256) {
            const int r  = idx >> 3;
            const int c4 = (idx & 7) * 4;
            *(float4*)&sAa[r * LDK + c4] = *(const float4*)&A [(size_t)(i0 + r) * DDIM + k0 + c4];
            *(float4*)&sAg[r * LDK + c4] = *(const float4*)&Sx[(size_t)(i0 + r) * DDIM + k0 + c4];
            *(float4*)&sBa[r * LDK + c4] = *(const float4*)&B [(size_t)(j0 + r) * DDIM + k0 + c4];
            *(float4*)&sBg[r * LDK + c4] = *(const float4*)&Sy[(size_t)(j0 + r) * DDIM + k0 + c4];
        }
        // warm the next K-chunk (global_prefetch_b8); operands are L2-resident
        if (k0 + KC < DDIM) {
            const int r = tid & 127;
            if (tid < 128) {
                __builtin_prefetch(&A [(size_t)(i0 + r) * DDIM + k0 + KC], 0, 3);
                __builtin_prefetch(&Sx[(size_t)(i0 + r) * DDIM + k0 + KC], 0, 3);
            } else {
                __builtin_prefetch(&B [(size_t)(j0 + r) * DDIM + k0 + KC], 0, 3);
                __builtin_prefetch(&Sy[(size_t)(j0 + r) * DDIM + k0 + KC], 0, 3);
            }
        }
        __syncthreads();

#pragma unroll
        for (int kk = 0; kk < KC; kk += 4) {
            const int ksel = kk + 2 * half;           // even dword -> 8B aligned
            v2f aA[2], aG[2], bA[4], bG[4];
#pragma unroll
            for (int ti = 0; ti < 2; ++ti) {
                const int ra = (wi * 32 + ti * 16 + lj) * LDK + ksel;
                aA[ti] = *(const v2f*)(sAa + ra);
                aG[ti] = *(const v2f*)(sAg + ra);
            }
#pragma unroll
            for (int tj = 0; tj < 4; ++tj) {
                const int rb = (wj * 64 + tj * 16 + lj) * LDK + ksel;
                bA[tj] = *(const v2f*)(sBa + rb);
                bG[tj] = *(const v2f*)(sBg + rb);
            }
#pragma unroll
            for (int ti = 0; ti < 2; ++ti)
#pragma unroll
                for (int tj = 0; tj < 4; ++tj) {
                    cA[ti][tj] = __builtin_amdgcn_wmma_f32_16x16x4_f32(
                        false, aA[ti], false, bA[tj], (short)0, cA[ti][tj], false, false);
                    cG[ti][tj] = __builtin_amdgcn_wmma_f32_16x16x4_f32(
                        false, aG[ti], false, bG[tj], (short)0, cG[ti][tj], false, false);
                }
        }
    }

    // Epilogue: d2 = max(||a||^2 + ||b||^2 - 2ab, 0); blend adaptive/gaussian.
#pragma unroll
    for (int ti = 0; ti < 2; ++ti) {
#pragma unroll
        for (int tj = 0; tj < 4; ++tj) {
#pragma unroll
            for (int v = 0; v < 8; ++v) {
                const int il = wi * 32 + ti * 16 + v + 8 * half;
                const int jl = wj * 64 + tj * 16 + lj;
                const float ab = cA[ti][tj][v];
                float d2 = s_nA[il] + s_nB[jl] - 2.f * ab;
                d2 = fmaxf(d2, 0.f);
                const float ex = __expf(-0.5f * d2);
                const float co = __cosf(s_xb[il] - s_yb[jl]);
                float g2 = s_nSx[il] + s_nSy[jl] - 2.f * cG[ti][tj][v];
                g2 = fmaxf(g2, 0.f);
                const float gs = __expf(-0.5f * g2);
                out[(size_t)(i0 + il) * MROWS + (j0 + jl)] = 0.9f * (co * ex) + 0.1f * gs;
            }
        }
    }
}

extern "C" void kernel_launch(void* const* d_in, const int* in_sizes, int n_in,
                              void* d_out, int out_size, void* d_ws, size_t ws_size,
                              hipStream_t stream) {
    const float* X  = (const float*)d_in[0];
    const float* Y  = (const float*)d_in[1];
    const float* W  = (const float*)d_in[2];
    const float* b  = (const float*)d_in[3];
    const float* ls = (const float*)d_in[4];
    float* out = (float*)d_out;

    // workspace layout (floats)
    float* ws = (float*)d_ws;
    const size_t mat = (size_t)NROWS * DDIM;   // 2M floats each
    float* A   = ws;             // X@W
    float* B   = A  + mat;       // Y@W
    float* Sx  = B  + mat;       // X/ls
    float* Sy  = Sx + mat;       // Y/ls
    float* nA  = Sy + mat;
    float* nB  = nA + NROWS;
    float* nSx = nB + MROWS;
    float* nSy = nSx + NROWS;
    float* xb  = nSy + MROWS;
    float* yb  = xb + NROWS;

    precompute_side<<<NROWS, 256, 0, stream>>>(X, W, b, ls, A, Sx, nA, nSx, xb);
    precompute_side<<<MROWS, 256, 0, stream>>>(Y, W, b, ls, B, Sy, nB, nSy, yb);

    dim3 grid(MROWS / 128, NROWS / 128);   // x: j tiles, y: i tiles
    bochner_main<<<grid, 256, 0, stream>>>(A, B, Sx, Sy, nA, nB, nSx, nSy, xb, yb, out);
}